// Model_22265110462511
// MI455X (gfx1250) — compile-verified
//
#include <hip/hip_runtime.h>
#include <stdint.h>

#define N_  65536
#define D_  1024
#define M_  32768
#define ND  ((size_t)N_ * D_)

#define ROWS 4                    // rows of updates/indices per TDM tile
#define TILE_ELEMS (ROWS * D_)    // 4096 elements per tile

typedef __attribute__((ext_vector_type(4))) unsigned int u32x4;
typedef __attribute__((ext_vector_type(8))) int          i32x8;
typedef __attribute__((ext_vector_type(4))) int          i32x4;
typedef __attribute__((ext_vector_type(4))) float        f32x4;

// ---------------------------------------------------------------------------
// TDM: issue a 2D tile load (rows x width elements, row stride = width) from
// global memory into LDS via the Tensor Data Mover. D# packed per CDNA5 ISA
// 8.3/8.4. data_size_code: 0=1B,1=2B,2=4B,3=8B.
// ---------------------------------------------------------------------------
__device__ __forceinline__ void tdm_load_2d(void* lds_ptr, const void* gptr,
                                            int data_size_code, int width, int rows) {
    uint64_t ga       = (uint64_t)(uintptr_t)gptr;
    uint32_t lds_addr = (uint32_t)(uintptr_t)lds_ptr;   // LDS aperture low bits = LDS byte offset
    unsigned w = (unsigned)width, r = (unsigned)rows;

    // Group 0 (128b): [1:0]=count=1, [63:32]=lds_addr, [120:64]=global_addr, [127:126]=type=2
    u32x4 g0 = { 1u,
                 lds_addr,
                 (uint32_t)ga,
                 (uint32_t)((ga >> 32) & 0x1FFFFFFull) | 0x80000000u };

    // Group 1 (256b):
    //  w0: workgroup_mask=0, [17:16]=data_size, flags=0
    //  [79:48]=tensor_dim0, [111:80]=tensor_dim1, [127:112]=tile_dim0,
    //  [143:128]=tile_dim1, [159:144]=tile_dim2=0, [207:160]=tensor_dim0_stride,
    //  [255:208]=tensor_dim1_stride (ignored for 2D here)
    i32x8 g1 = { (int)((unsigned)data_size_code << 16),
                 (int)((w & 0xFFFFu) << 16),                          // tensor_dim0[15:0]
                 (int)(((w >> 16) & 0xFFFFu) | ((r & 0xFFFFu) << 16)),// td0[31:16] | td1[15:0]
                 (int)((w & 0xFFFFu) << 16),                          // td1[31:16]=0 | tile_dim0
                 (int)(r & 0xFFFFu),                                  // tile_dim1 | tile_dim2=0
                 (int)w,                                              // tensor_dim0_stride[31:0]
                 0, 0 };

    i32x4 gz4 = {0, 0, 0, 0};                 // groups 2/3 unused for 2D
    i32x8 gz8 = {0, 0, 0, 0, 0, 0, 0, 0};     // 5th group (clang-23 6-arg form)
    __builtin_amdgcn_tensor_load_to_lds(g0, g1, gz4, gz4, gz8, 0);
}

// ---------------------------------------------------------------------------
// Kernel 1: out = float(var_int8) * var_scale   (also zero the ws abs-max slot)
// 16 elements / thread: 16B NT load, 64B store.
// ---------------------------------------------------------------------------
__global__ __launch_bounds__(256)
void k_dequant_init(const int8_t* __restrict__ var, const float* __restrict__ var_scale,
                    float* __restrict__ outf, unsigned int* __restrict__ ws_max) {
    if (blockIdx.x == 0 && threadIdx.x == 0) *ws_max = 0u;
    const float s = *var_scale;
    size_t base = ((size_t)blockIdx.x * 256 + threadIdx.x) * 16;
    i32x4 wv = __builtin_nontemporal_load((const i32x4*)(var + base));
    int words[4] = {wv.x, wv.y, wv.z, wv.w};
    f32x4* op = (f32x4*)(outf + base);
#pragma unroll
    for (int q = 0; q < 4; ++q) {
        int v = words[q];
        f32x4 o;
        o.x = (float)(int8_t)(v)       * s;
        o.y = (float)(int8_t)(v >> 8)  * s;
        o.z = (float)(int8_t)(v >> 16) * s;
        o.w = (float)(int8_t)(v >> 24) * s;
        op[q] = o;
    }
}

// ---------------------------------------------------------------------------
// Kernel 2: scatter-add with TDM double buffering.
// Block: 256 threads (8 waves). Wave 0 issues TDM tile DMAs (indices+updates)
// into the back LDS buffer, s_wait_tensorcnt gates the front buffer, then all
// waves drain 16 non-returning global_atomic_add_f32 per thread per tile.
// ---------------------------------------------------------------------------
__global__ __launch_bounds__(256)
void k_scatter(const int* __restrict__ indices, const _Float16* __restrict__ updates,
               const _Float16* __restrict__ smooth, float* __restrict__ outf) {
    __shared__ alignas(16) int       sh_idx[2][TILE_ELEMS];   // 2 x 16 KB
    __shared__ alignas(16) _Float16  sh_upd[2][TILE_ELEMS];   // 2 x  8 KB
    __shared__ alignas(16) _Float16  sh_sc[D_];               //      2 KB

    // Stage smooth_scales once: 1024 halfs = 256 x 8B
    ((uint64_t*)sh_sc)[threadIdx.x] = ((const uint64_t*)smooth)[threadIdx.x];

    const int  tiles_total = M_ / ROWS;              // 8192
    const bool wave0 = (threadIdx.x < 32);           // wave-uniform gate
    int  tile = blockIdx.x;
    int  buf  = 0;

    if (wave0 && tile < tiles_total) {
        size_t r0 = (size_t)tile * ROWS;
        tdm_load_2d(&sh_idx[0][0], indices + r0 * D_, 2, D_, ROWS);
        tdm_load_2d(&sh_upd[0][0], updates + r0 * D_, 1, D_, ROWS);
    }

    for (; tile < tiles_total; tile += gridDim.x) {
        const int next = tile + gridDim.x;
        if (wave0) {
            if (next < tiles_total) {
                size_t r0 = (size_t)next * ROWS;
                tdm_load_2d(&sh_idx[buf ^ 1][0], indices + r0 * D_, 2, D_, ROWS);
                tdm_load_2d(&sh_upd[buf ^ 1][0], updates + r0 * D_, 1, D_, ROWS);
                __builtin_amdgcn_s_wait_tensorcnt(2);  // current tile's 2 DMAs done
            } else {
                __builtin_amdgcn_s_wait_tensorcnt(0);
            }
        }
        __syncthreads();                               // publish front buffer

        // 4096 elems / 256 threads = 16 per thread, contiguous
        const int e0 = threadIdx.x * 16;
#pragma unroll
        for (int u = 0; u < 16; u += 4) {
            const int e = e0 + u;
            const int j = e & (D_ - 1);
            i32x4 idx4 = *(const i32x4*)&sh_idx[buf][e];
            const int* ip = (const int*)&idx4;
#pragma unroll
            for (int k = 0; k < 4; ++k) {
                float up = (float)sh_upd[buf][e + k] * (float)sh_sc[j + k];
                float* p = outf + (size_t)ip[k] * D_ + (size_t)(j + k);
                unsafeAtomicAdd(p, up);                // no-return global_atomic_add_f32
            }
        }
        __syncthreads();                               // front buffer consumed
        buf ^= 1;
    }
}

// ---------------------------------------------------------------------------
// Kernel 3: global abs-max over out (64M floats), 16 / thread, wave shfl_xor
// reduce + LDS + one u32 atomicMax per block (abs-float bits are monotonic).
// ---------------------------------------------------------------------------
__global__ __launch_bounds__(256)
void k_absmax(const float* __restrict__ outf, unsigned int* __restrict__ ws_max) {
    size_t base = ((size_t)blockIdx.x * 256 + threadIdx.x) * 16;
    float m = 0.0f;
#pragma unroll
    for (int q = 0; q < 4; ++q) {
        f32x4 v = __builtin_nontemporal_load((const f32x4*)(outf + base + q * 4));
        m = fmaxf(m, fmaxf(fmaxf(fabsf(v.x), fabsf(v.y)), fmaxf(fabsf(v.z), fabsf(v.w))));
    }
#pragma unroll
    for (int off = 16; off > 0; off >>= 1)
        m = fmaxf(m, __shfl_xor(m, off, 32));
    __shared__ float sm[8];
    if ((threadIdx.x & 31) == 0) sm[threadIdx.x >> 5] = m;
    __syncthreads();
    if (threadIdx.x == 0) {
        float mm = sm[0];
#pragma unroll
        for (int w = 1; w < 8; ++w) mm = fmaxf(mm, sm[w]);
        atomicMax(ws_max, __float_as_uint(mm));
    }
}

// ---------------------------------------------------------------------------
// Kernel 4: y = clip(rint(out / new_scale), -128, 127) (as float), + new_scale
// ---------------------------------------------------------------------------
__global__ __launch_bounds__(256)
void k_quant(const float* __restrict__ outf, const unsigned int* __restrict__ ws_max,
             float* __restrict__ y, float* __restrict__ scale_out) {
    const float mx = __uint_as_float(*ws_max);
    const float ns = mx * (1.0f / 127.0f);
    const float inv = (ns > 0.0f) ? 1.0f / ns : 0.0f;
    size_t base = ((size_t)blockIdx.x * 256 + threadIdx.x) * 8;
#pragma unroll
    for (int q = 0; q < 2; ++q) {
        f32x4 v = __builtin_nontemporal_load((const f32x4*)(outf + base + q * 4));
        f32x4 r;
        r.x = fminf(fmaxf(rintf(v.x * inv), -128.0f), 127.0f);
        r.y = fminf(fmaxf(rintf(v.y * inv), -128.0f), 127.0f);
        r.z = fminf(fmaxf(rintf(v.z * inv), -128.0f), 127.0f);
        r.w = fminf(fmaxf(rintf(v.w * inv), -128.0f), 127.0f);
        __builtin_nontemporal_store(r, (f32x4*)(y + base + q * 4));
    }
    if (blockIdx.x == 0 && threadIdx.x == 0) *scale_out = ns;
}

// ---------------------------------------------------------------------------
extern "C" void kernel_launch(void* const* d_in, const int* in_sizes, int n_in,
                              void* d_out, int out_size, void* d_ws, size_t ws_size,
                              hipStream_t stream) {
    const int8_t*   var       = (const int8_t*)  d_in[0];
    const float*    var_scale = (const float*)   d_in[1];
    const int*      indices   = (const int*)     d_in[2];
    const _Float16* updates   = (const _Float16*)d_in[3];
    const _Float16* smooth    = (const _Float16*)d_in[4];

    float* y         = (float*)d_out;          // tuple slot 0: y (as float)
    float* outf      = y + ND;                 // tuple slot 1: out (fp32) — used as accumulator
    float* scale_out = y + 2 * ND;             // tuple slot 2: new_scale
    unsigned int* ws_max = (unsigned int*)d_ws;

    // 1) out = dequant(var); zero abs-max slot
    k_dequant_init<<<(unsigned)(ND / (256 * 16)), 256, 0, stream>>>(var, var_scale, outf, ws_max);
    // 2) TDM double-buffered scatter-add (32M fp32 atomics)
    k_scatter<<<2048, 256, 0, stream>>>(indices, updates, smooth, outf);
    // 3) global abs-max
    k_absmax<<<(unsigned)(ND / (256 * 16)), 256, 0, stream>>>(outf, ws_max);
    // 4) requantize + emit scale
    k_quant<<<(unsigned)(ND / (256 * 8)), 256, 0, stream>>>(outf, ws_max, y, scale_out);
}